// RandomTopKGate_73134703116977
// MI455X (gfx1250) — compile-verified
//
#include <hip/hip_runtime.h>
#include <cstdint>

#define NE 64            // experts per row (256 B per row)
#define RPB 32           // rows per block == threads per block (single wave32)
#define CHUNKS 16        // 16-byte chunks per row
#define LSTRIDE 68       // LDS floats per row (272 B = 256 + 16 pad -> 2-way banks)

// med3(x,y,v) with invariant x >= y: one v_med3_num_f32 (or min/max fallback)
#if defined(__has_builtin) && __has_builtin(__builtin_amdgcn_fmed3f)
#define MED3(x, y, v) __builtin_amdgcn_fmed3f((x), (y), (v))
#else
#define MED3(x, y, v) fminf((x), fmaxf((y), (v)))
#endif

// Branchless insert of v into descending triple (a1 >= a2 >= a3): 3 VALU.
__device__ __forceinline__ void insert3(float& a1, float& a2, float& a3, float v) {
    const float o1 = a1, o2 = a2;
    a1 = fmaxf(o1, v);
    a2 = MED3(o1, o2, v);
    a3 = MED3(o2, a3, v);
}

__global__ __launch_bounds__(RPB)
void topk_gate_kernel(const float* __restrict__ routing,
                      float* __restrict__ out, int tokens) {
    __shared__ float tile[RPB * LSTRIDE];   // 8.5 KB padded staging

    const unsigned t = threadIdx.x;         // 0..31 ; lane == thread == local row
    const int row0 = (int)(blockIdx.x * RPB);

    if (row0 + RPB <= tokens) {
        const float* gin  = routing + (size_t)row0 * NE;
        float*       gout = out     + (size_t)row0 * NE;

        const unsigned lbase = (unsigned)(uintptr_t)(&tile[0]);
        // iter k: global bytes [k*512 + t*16, +16) == row r=(t>>4)+2k, chunk j=t&15
        // LDS addr = lbase + r*272 + j*16 = l0 + 544k; shared imm gives +512k,
        // so the LDS VGPR carries l0 + 32k (one v_add per issue).
        const unsigned l0    = lbase + (t >> 4) * 272u + (t & 15u) * 16u;
        const unsigned g_off = t * 16u;

        // ---- CDNA5 async coalesced global->LDS (ASYNCcnt path) -------------
        #pragma unroll
        for (unsigned k = 0; k < CHUNKS; ++k) {
            asm volatile("global_load_async_to_lds_b128 %0, %1, %2 offset:%3"
                         :: "v"(l0 + 32u * k), "v"(g_off), "s"(gin), "i"(k * 512)
                         : "memory");
        }
        asm volatile("s_wait_asynccnt 0" ::: "memory");
        // single-wave workgroup: ASYNCcnt==0 makes LDS visible, no barrier

        // ---- per-thread top-3 over own row; DS reads: base + imm offsets ---
        const float* rowp = &tile[t * LSTRIDE];
        float4 c[CHUNKS];
        float a1 = -__builtin_inff(), a2 = a1, a3 = a1;   // chain A: x,y
        float b1 = a1, b2 = a1, b3 = a1;                  // chain B: z,w
        #pragma unroll
        for (unsigned j = 0; j < CHUNKS; ++j) {
            c[j] = *reinterpret_cast<const float4*>(rowp + j * 4u);
            insert3(a1, a2, a3, c[j].x);
            insert3(b1, b2, b3, c[j].z);
            insert3(a1, a2, a3, c[j].y);
            insert3(b1, b2, b3, c[j].w);
        }
        // merge sorted triple B into A (b1 >= b2 >= b3): 3 + 2 + 2 ops
        insert3(a1, a2, a3, b1);
        {
            float n2 = fmaxf(a2, b2); a3 = MED3(a2, a3, b2); a2 = n2;   // b2 <= a1
            n2       = fmaxf(a2, b3); a3 = MED3(a2, a3, b3); a2 = n2;   // b3 <= a1
        }

        // quantile(1 - 2/64): index 61.03125 ascending -> lerp(3rd, 2nd, 1/32)
        const float quant = a3 * 0.96875f + a2 * 0.03125f;
        // round(m/(m+0.01)) == (m > 0.01) for kept values; fold into one gate
        const float gate = fmaxf(quant, 0.01f);

        // ---- stage gates back to LDS (same base + imm offsets) -------------
        float* roww = &tile[t * LSTRIDE];
        #pragma unroll
        for (unsigned j = 0; j < CHUNKS; ++j) {
            float4 o;
            o.x = (c[j].x > gate) ? 1.0f : 0.0f;
            o.y = (c[j].y > gate) ? 1.0f : 0.0f;
            o.z = (c[j].z > gate) ? 1.0f : 0.0f;
            o.w = (c[j].w > gate) ? 1.0f : 0.0f;
            *reinterpret_cast<float4*>(roww + j * 4u) = o;
        }
        asm volatile("s_wait_dscnt 0" ::: "memory");  // LDS visible to async engine

        // ---- CDNA5 async coalesced LDS->global stores ----------------------
        #pragma unroll
        for (unsigned k = 0; k < CHUNKS; ++k) {
            asm volatile("global_store_async_from_lds_b128 %0, %1, %2 offset:%3"
                         :: "v"(g_off), "v"(l0 + 32u * k), "s"(gout), "i"(k * 512)
                         : "memory");
        }
        asm volatile("s_wait_asynccnt 0" ::: "memory");
    } else {
        // remainder rows (tokens % 32): direct per-thread path
        const int row = row0 + (int)t;
        if (row < tokens) {
            const float* rp = routing + (size_t)row * NE;
            float a1 = -__builtin_inff(), a2 = a1, a3 = a1;
            float4 c[CHUNKS];
            #pragma unroll
            for (unsigned j = 0; j < CHUNKS; ++j) {
                c[j] = *reinterpret_cast<const float4*>(rp + j * 4u);
                insert3(a1, a2, a3, c[j].x);
                insert3(a1, a2, a3, c[j].y);
                insert3(a1, a2, a3, c[j].z);
                insert3(a1, a2, a3, c[j].w);
            }
            const float gate = fmaxf(a3 * 0.96875f + a2 * 0.03125f, 0.01f);
            float* op = out + (size_t)row * NE;
            #pragma unroll
            for (unsigned j = 0; j < CHUNKS; ++j) {
                float4 o;
                o.x = (c[j].x > gate) ? 1.0f : 0.0f;
                o.y = (c[j].y > gate) ? 1.0f : 0.0f;
                o.z = (c[j].z > gate) ? 1.0f : 0.0f;
                o.w = (c[j].w > gate) ? 1.0f : 0.0f;
                *reinterpret_cast<float4*>(op + j * 4u) = o;
            }
        }
    }
}

extern "C" void kernel_launch(void* const* d_in, const int* in_sizes, int n_in,
                              void* d_out, int out_size, void* d_ws, size_t ws_size,
                              hipStream_t stream) {
    // d_in[0] = input [TOKENS, 2048] f32 -- unused by the reference math.
    // d_in[1] = routing [TOKENS, 64] f32.
    const float* routing = (const float*)d_in[1];
    float* out = (float*)d_out;

    const int tokens = out_size / NE;                 // 32768
    const int grid = (tokens + RPB - 1) / RPB;        // 1024 single-wave blocks

    topk_gate_kernel<<<grid, RPB, 0, stream>>>(routing, out, tokens);
}